// GraphLossLayerPairs_996432413471
// MI455X (gfx1250) — compile-verified
//
#include <hip/hip_runtime.h>
#include <hip/hip_bf16.h>

typedef __attribute__((ext_vector_type(2))) float v2f;
typedef __attribute__((ext_vector_type(8))) float v8f;
typedef __attribute__((ext_vector_type(2))) long long v2ll;

// Exact f32 wave(32)-wide sum using V_WMMA_F32_16X16X4_F32.
// A (16x4 f32, 2 VGPRs): lane m holds A[m][0] (VGPR0 lanes 0-15), lane m+16
// holds A[m][2] (VGPR0 lanes 16-31); VGPR1 (K=1,3) set to zero.
// B = all ones  =>  D[m][n] = A[m][0]+A[m][2] = acc[m] + acc[m+16].
// Per-lane sum of the 8 D VGPRs gives rows 0-7 (lanes 0-15) / rows 8-15
// (lanes 16-31); one shfl_xor(16) completes the 32-lane sum in every lane.
__device__ __forceinline__ float wave_sum_wmma(float v) {
    v2f a; a[0] = v;    a[1] = 0.0f;
    v2f b; b[0] = 1.0f; b[1] = 1.0f;
    v8f c = {};
    v8f d = __builtin_amdgcn_wmma_f32_16x16x4_f32(
        /*neg_a=*/false, a, /*neg_b=*/false, b,
        /*c_mod=*/(short)0, c, /*reuse_a=*/false, /*reuse_b=*/false);
    float s = ((d[0] + d[1]) + (d[2] + d[3])) + ((d[4] + d[5]) + (d[6] + d[7]));
    s += __shfl_xor(s, 16, 32);
    return s;
}

// Deterministic block-wide sum (block must be a multiple of 32, <= 256).
__device__ __forceinline__ float block_sum(float v, float* lds) {
    const int lane = threadIdx.x & 31;
    const int wid  = threadIdx.x >> 5;
    const int nw   = blockDim.x >> 5;
    float w = wave_sum_wmma(v);
    __syncthreads();                 // protect lds reuse across calls
    if (lane == 0) lds[wid] = w;
    __syncthreads();
    if (wid == 0) {                  // wave 0 executes with full EXEC
        float t = (lane < nw) ? lds[lane] : 0.0f;
        float r = wave_sum_wmma(t);
        if (lane == 0) lds[0] = r;
    }
    __syncthreads();
    return lds[0];
}

struct F3 { float x, y, z; };        // 12-byte point -> global_load_b96

// ---------------- Kernel 1: distance-field loss  sum(d^2) ----------------
__global__ void __launch_bounds__(256)
dist_loss_kernel(const float* __restrict__ V, int N,
                 const float* __restrict__ F,   // 64^3, idx = (x*64+y)*64+z
                 float* __restrict__ partial) {
    __shared__ float lds[8];
    float acc = 0.0f;
    const int stride = gridDim.x * blockDim.x;
    const F3* __restrict__ V3 = (const F3*)V;
    for (int i = blockIdx.x * blockDim.x + threadIdx.x; i < N; i += stride) {
        F3 p = V3[i];
        float cx = fminf(fmaxf(p.x * 63.0f, 0.0f), 62.9999f);
        float cy = fminf(fmaxf(p.y * 63.0f, 0.0f), 62.9999f);
        float cz = fminf(fmaxf(p.z * 63.0f, 0.0f), 62.9999f);
        int x0 = (int)cx, y0 = (int)cy, z0 = (int)cz;   // floor (non-negative)
        float fx = cx - (float)x0, fy = cy - (float)y0, fz = cz - (float)z0;
        int x1 = min(x0 + 1, 63), y1 = min(y0 + 1, 63), z1 = min(z0 + 1, 63);
        const float* Fx0 = F + x0 * 4096;
        const float* Fx1 = F + x1 * 4096;
        int y0o = y0 * 64, y1o = y1 * 64;
        float c000 = Fx0[y0o + z0], c001 = Fx0[y0o + z1];
        float c010 = Fx0[y1o + z0], c011 = Fx0[y1o + z1];
        float c100 = Fx1[y0o + z0], c101 = Fx1[y0o + z1];
        float c110 = Fx1[y1o + z0], c111 = Fx1[y1o + z1];
        float omz = 1.0f - fz, omy = 1.0f - fy, omx = 1.0f - fx;
        float c00 = c000 * omz + c001 * fz;
        float c01 = c010 * omz + c011 * fz;
        float c10 = c100 * omz + c101 * fz;
        float c11 = c110 * omz + c111 * fz;
        float c0  = c00 * omy + c01 * fy;
        float c1  = c10 * omy + c11 * fy;
        float d   = c0 * omx + c1 * fx;
        acc += d * d;
    }
    float bs = block_sum(acc, lds);
    if (threadIdx.x == 0) partial[blockIdx.x] = bs;
}

// ---------------- Kernel 2: edge rigidity loss  sum(|diff|^2) -------------
// E (192 MB) is streamed once -> non-temporal B128 loads so it does not
// evict the V/V_ref working set (48 MB) that the gathers re-hit in L2.
__global__ void __launch_bounds__(256)
edge_loss_kernel(const float* __restrict__ V, const float* __restrict__ Vr,
                 const long long* __restrict__ E, int M,
                 float* __restrict__ partial) {
    __shared__ float lds[8];
    float acc = 0.0f;
    const int stride = gridDim.x * blockDim.x;
    const F3*   __restrict__ V3  = (const F3*)V;
    const F3*   __restrict__ Vr3 = (const F3*)Vr;
    const v2ll* __restrict__ E2  = (const v2ll*)E;     // rows are 16B aligned
    for (int i = blockIdx.x * blockDim.x + threadIdx.x; i < M; i += stride) {
        v2ll e = __builtin_nontemporal_load(&E2[i]);   // global_load_b128 NT
        long long s = e[0];
        long long t = e[1];
        F3 vs = V3[s],  vt = V3[t];
        F3 rs = Vr3[s], rt = Vr3[t];
        float dx = (vs.x - vt.x) - (rs.x - rt.x);
        float dy = (vs.y - vt.y) - (rs.y - rt.y);
        float dz = (vs.z - vt.z) - (rs.z - rt.z);
        acc += dx * dx + dy * dy + dz * dz;
    }
    float bs = block_sum(acc, lds);
    if (threadIdx.x == 0) partial[blockIdx.x] = bs;
}

// ---------------- Kernel 3: deterministic finalize ------------------------
__global__ void __launch_bounds__(256)
finalize_kernel(const float* __restrict__ pd, int nd,
                const float* __restrict__ pe, int ne,
                const float* __restrict__ r2, float* __restrict__ out) {
    __shared__ float lds[8];
    float a = 0.0f;
    for (int i = threadIdx.x; i < nd; i += blockDim.x) a += pd[i];
    float lossD = block_sum(a, lds);
    float b = 0.0f;
    for (int i = threadIdx.x; i < ne; i += blockDim.x) b += pe[i];
    float lossR = block_sum(b, lds);
    if (threadIdx.x == 0) out[0] = 0.5f * lossD + 0.5f * lossR * r2[0];
}

extern "C" void kernel_launch(void* const* d_in, const int* in_sizes, int n_in,
                              void* d_out, int out_size, void* d_ws, size_t ws_size,
                              hipStream_t stream) {
    const float*     V  = (const float*)d_in[0];
    const long long* E  = (const long long*)d_in[1];   // int64 [M,2]
    const float*     Vr = (const float*)d_in[2];
    const float*     F  = (const float*)d_in[3];       // [64,64,64]
    const float*     r2 = (const float*)d_in[4];       // scalar

    const int N = in_sizes[0] / 3;
    const int M = in_sizes[1] / 2;

    const int BS = 256;
    const int GD = 1024;   // dist partials
    const int GE = 2048;   // edge partials

    float* pd = (float*)d_ws;
    float* pe = pd + GD;   // 12 KB of ws used; every slot written each call

    dist_loss_kernel<<<GD, BS, 0, stream>>>(V, N, F, pd);
    edge_loss_kernel<<<GE, BS, 0, stream>>>(V, Vr, E, M, pe);
    finalize_kernel<<<1, BS, 0, stream>>>(pd, GD, pe, GE, r2, (float*)d_out);
}